// APPM_77481210020195
// MI455X (gfx1250) — compile-verified
//
#include <hip/hip_runtime.h>
#include <math.h>

#define FEAT 112
#define TOTAL_W 96981
#define NRATIO 13
#define PROW 116            // padded prefix row stride (floats); rows 16B-aligned
#define PIMG 13120          // per-image prefix stride (floats), >= 113*116
#define IDX_OFF 0           // d_out: indices (256*7 floats)
#define SCR_OFF 1792        // d_out: picked scores (256*7 floats)
#define WSC_OFF 3584        // d_out: window scores (256*96981 floats)

typedef __attribute__((ext_vector_type(2))) float v2f;
typedef __attribute__((ext_vector_type(8))) float v8f;

// Types matching the async-to-LDS builtin's parameters:
//   void(int4 addrspace(1)*, int4 addrspace(3)*, imm int offset, imm int cpol)
typedef int v4i_t __attribute__((vector_size(4 * sizeof(int))));
typedef __attribute__((address_space(1))) v4i_t* g_v4i_p;
typedef __attribute__((address_space(3))) v4i_t* l_v4i_p;

__constant__ int c_RH[NRATIO]    = {16,12,20,24,20,28,32,24,40,28,40,28,36};
__constant__ int c_RW[NRATIO]    = {16,20,12,24,28,20,32,40,24,40,28,36,28};
__constant__ int c_OFF[NRATIO+1] = {0,9409,18802,28195,36116,44021,51926,
                                    58487,64984,71481,77686,83891,90436,96981};

#if defined(__AMDGCN__) && __has_builtin(__builtin_amdgcn_wmma_f32_16x16x4_f32)
#define HAVE_WMMA_F32 1
#else
#define HAVE_WMMA_F32 0
#endif

// Low 32 bits of a generic LDS pointer are the LDS byte offset (flat LDS
// addressing, ISA 10.2).
__device__ __forceinline__ unsigned lds_u32(const void* p) {
  return (unsigned)(size_t)p;
}

__device__ __forceinline__ void async_g2l_b128(const float* g, float* l) {
#if defined(__AMDGCN__) && __has_builtin(__builtin_amdgcn_global_load_async_to_lds_b128)
  __builtin_amdgcn_global_load_async_to_lds_b128(
      (g_v4i_p)(size_t)g, (l_v4i_p)(size_t)lds_u32(l), 0, 0);
#else
  asm volatile("global_load_async_to_lds_b128 %0, %1, off"
               :: "v"(lds_u32(l)), "v"(g) : "memory");
#endif
}

__device__ __forceinline__ void wait_async0() {
#if defined(__AMDGCN__) && __has_builtin(__builtin_amdgcn_s_wait_asynccnt)
  __builtin_amdgcn_s_wait_asynccnt(0);
#else
  asm volatile("s_wait_asynccnt 0" ::: "memory");
#endif
}

// ---------------------------------------------------------------------------
// Kernel 1: Y = L * X  (inclusive column prefix) via V_WMMA_F32_16X16X4_F32.
// One workgroup (256 thr = 8 waves) per image. X staged in LDS via async copy.
// Y stored at rows 0..111 / cols 0..111 of the padded prefix region.
// ---------------------------------------------------------------------------
__global__ __launch_bounds__(256) void k_colprefix(const float* __restrict__ x,
                                                   float* __restrict__ wsP) {
  __shared__ __attribute__((aligned(16))) float sX[FEAT * FEAT];
  const int tid = threadIdx.x;
  const int img = blockIdx.x;
  const float* gx = x + (size_t)img * (FEAT * FEAT);
  for (int c = tid; c < (FEAT * FEAT) / 4; c += 256)
    async_g2l_b128(gx + c * 4, &sX[c * 4]);
  wait_async0();
  __syncthreads();

  float* yb = wsP + (size_t)img * PIMG;
#if HAVE_WMMA_F32
  const int wave = tid >> 5, lane = tid & 31, hi = lane >> 4, lo = lane & 15;
  for (int t = wave; t < 49; t += 8) {            // wave-uniform: EXEC all ones
    const int it = t / 7, jt = t % 7;
    const int m0 = it * 16, j0 = jt * 16;
    const int m = m0 + lo;
    v8f acc = {0.f, 0.f, 0.f, 0.f, 0.f, 0.f, 0.f, 0.f};
    const int kmax = (it + 1) * 16;               // L[m,k]=0 for k>m
    for (int k = 0; k < kmax; k += 4) {
      const int kb = k + 2 * hi;
      v2f a, b;
      a.x = (kb     <= m) ? 1.0f : 0.0f;          // A = L tile chunk (generated)
      a.y = (kb + 1 <= m) ? 1.0f : 0.0f;
      b.x = sX[kb * FEAT + j0 + lo];              // B = X chunk (4 x 16)
      b.y = sX[(kb + 1) * FEAT + j0 + lo];
      acc = __builtin_amdgcn_wmma_f32_16x16x4_f32(false, a, false, b,
                                                  (short)0, acc, false, false);
    }
#pragma unroll
    for (int r = 0; r < 8; ++r)
      yb[(m0 + r + 8 * hi) * PROW + j0 + lo] = acc[r];
  }
#else
  for (int j = tid; j < FEAT; j += 256) {
    float s = 0.f;
    for (int i = 0; i < FEAT; ++i) { s += sX[i * FEAT + j]; yb[i * PROW + j] = s; }
  }
#endif
}

// ---------------------------------------------------------------------------
// Kernel 2: P = Y * U (inclusive row prefix) via WMMA; in-place into the
// padded region (Y fully staged in LDS first). Pads row 0 / col 0 with zeros,
// so stage-3 corner reads are branch-free.
// ---------------------------------------------------------------------------
__global__ __launch_bounds__(256) void k_rowprefix(float* __restrict__ wsP) {
  __shared__ __attribute__((aligned(16))) float sY[FEAT * FEAT];
  const int tid = threadIdx.x;
  const int img = blockIdx.x;
  float* pb = wsP + (size_t)img * PIMG;
  for (int c = tid; c < (FEAT * FEAT) / 4; c += 256) {
    const int row = c / 28, c4 = c % 28;          // 28 float4 per 112-float row
    async_g2l_b128(pb + row * PROW + c4 * 4, &sY[row * FEAT + c4 * 4]);
  }
  wait_async0();
  __syncthreads();

#if HAVE_WMMA_F32
  const int wave = tid >> 5, lane = tid & 31, hi = lane >> 4, lo = lane & 15;
  for (int t = wave; t < 49; t += 8) {
    const int it = t / 7, jt = t % 7;
    const int m0 = it * 16, j0 = jt * 16;
    v8f acc = {0.f, 0.f, 0.f, 0.f, 0.f, 0.f, 0.f, 0.f};
    const int kmax = (jt + 1) * 16;               // U[k,j]=0 for k>j
    for (int k = 0; k < kmax; k += 4) {
      const int kb = k + 2 * hi;
      v2f a, b;
      a.x = sY[(m0 + lo) * FEAT + kb];            // A = Y chunk (16 x 4)
      a.y = sY[(m0 + lo) * FEAT + kb + 1];
      b.x = (kb     <= j0 + lo) ? 1.0f : 0.0f;    // B = U tile chunk (generated)
      b.y = (kb + 1 <= j0 + lo) ? 1.0f : 0.0f;
      acc = __builtin_amdgcn_wmma_f32_16x16x4_f32(false, a, false, b,
                                                  (short)0, acc, false, false);
    }
#pragma unroll
    for (int r = 0; r < 8; ++r)
      pb[(m0 + r + 8 * hi + 1) * PROW + (j0 + lo + 1)] = acc[r];
  }
#else
  for (int i = tid; i < FEAT; i += 256) {
    float s = 0.f;
    for (int j = 0; j < FEAT; ++j) { s += sY[i * FEAT + j]; pb[(i + 1) * PROW + j + 1] = s; }
  }
#endif
  // zero padding borders
  for (int i = tid; i < FEAT + 1; i += 256) pb[i] = 0.f;              // row 0
  for (int i = tid; i < FEAT; i += 256)     pb[(i + 1) * PROW] = 0.f; // col 0
}

// ---------------------------------------------------------------------------
// Kernel 3: window scores via 4 integral-image corner reads (memory-bound;
// ~99MB of stores is the roofline for this problem).
// ---------------------------------------------------------------------------
__global__ __launch_bounds__(256) void k_winscores(const float* __restrict__ wsP,
                                                   float* __restrict__ out) {
  const int w = blockIdx.x * 256 + threadIdx.x;
  const int b = blockIdx.y;
  if (w >= TOTAL_W) return;
  int r = 0;
  while (w >= c_OFF[r + 1]) ++r;
  const int rh = c_RH[r], rw = c_RW[r];
  const int nc = FEAT - rw + 1;
  const int loc = w - c_OFF[r];
  const int xi = loc / nc, yi = loc % nc;
  const float* P = wsP + (size_t)b * PIMG;
  const float s = P[(xi + rh) * PROW + yi + rw] - P[xi * PROW + yi + rw]
                - P[(xi + rh) * PROW + yi]      + P[xi * PROW + yi];
  out[WSC_OFF + (size_t)b * TOTAL_W + w] = s / (float)(rh * rw);
}

// ---------------------------------------------------------------------------
// Kernel 4: per-(batch, group) NMS. Block-wide argmax with reference-exact
// tie-break (lowest index), LDS alive-bitmask, IoU suppression.
// ---------------------------------------------------------------------------
__global__ __launch_bounds__(256) void k_nms(const float* __restrict__ coords,
                                             float* __restrict__ out) {
  __shared__ unsigned alive[1408];                 // 45056 bits >= max group size
  __shared__ float rv[256];
  __shared__ int   ri[256];
  __shared__ float sbox[4];
  __shared__ int   spick, salive;

  const int G_START[3] = {0, 28195, 51926};
  const int G_SIZE[3]  = {28195, 23731, 45055};
  const int G_NPICK[3] = {2, 3, 2};
  const int G_OUT[3]   = {0, 2, 5};

  const int tid = threadIdx.x;
  const int b = blockIdx.x, g = blockIdx.y;
  const int s0 = G_START[g], sz = G_SIZE[g], np = G_NPICK[g];
  const float* sc = out + WSC_OFF + (size_t)b * TOTAL_W + s0;
  const float* cd = coords + (size_t)4 * s0;
  const int nw = (sz + 31) >> 5;
  for (int i = tid; i < nw; i += 256) alive[i] = 0xFFFFFFFFu;
  __syncthreads();

  int last = 0;                                    // only thread 0 uses this
  for (int p = 0; p < np; ++p) {
    float bv = -INFINITY; int bi = 0x7FFFFFFF;
    for (int i = tid; i < sz; i += 256) {
      if ((alive[i >> 5] >> (i & 31)) & 1u) {
        const float s = sc[i];
        if (s > bv || (s == bv && i < bi)) { bv = s; bi = i; }
      }
    }
    rv[tid] = bv; ri[tid] = bi;
    __syncthreads();
    for (int off = 128; off > 0; off >>= 1) {
      if (tid < off) {
        const float v2 = rv[tid + off]; const int i2 = ri[tid + off];
        if (v2 > rv[tid] || (v2 == rv[tid] && i2 < ri[tid])) { rv[tid] = v2; ri[tid] = i2; }
      }
      __syncthreads();
    }
    if (tid == 0) {
      int pick;
      if (rv[0] == -INFINITY) { pick = last; salive = 0; }
      else                    { pick = ri[0]; salive = 1; }
      spick = pick; last = pick;
      sbox[0] = cd[4 * pick + 0]; sbox[1] = cd[4 * pick + 1];
      sbox[2] = cd[4 * pick + 2]; sbox[3] = cd[4 * pick + 3];
      out[IDX_OFF + (size_t)b * 7 + G_OUT[g] + p] = (float)(pick + s0);
      out[SCR_OFF + (size_t)b * 7 + G_OUT[g] + p] = sc[pick];
    }
    __syncthreads();
    if (salive) {
      const float ax0 = sbox[0], ay0 = sbox[1], ax1 = sbox[2], ay1 = sbox[3];
      const float pa = (ax1 - ax0 + 1.f) * (ay1 - ay0 + 1.f);
      for (int i = tid; i < sz; i += 256) {
        if ((alive[i >> 5] >> (i & 31)) & 1u) {
          const float x0 = cd[4 * i], y0 = cd[4 * i + 1];
          const float x1 = cd[4 * i + 2], y1 = cd[4 * i + 3];
          const float ltx = fmaxf(x0, ax0), lty = fmaxf(y0, ay0);
          const float rbx = fminf(x1, ax1), rby = fminf(y1, ay1);
          const float ww = rbx - ltx + 1.f, hh = rby - lty + 1.f;
          const float inter = (ww < 0.f || hh < 0.f) ? 0.f : ww * hh;
          const float ia = (x1 - x0 + 1.f) * (y1 - y0 + 1.f);
          const float iou = inter / (ia + pa - inter);
          if (iou > 0.25f) atomicAnd(&alive[i >> 5], ~(1u << (i & 31)));
        }
      }
    }
    __syncthreads();
  }
}

extern "C" void kernel_launch(void* const* d_in, const int* in_sizes, int n_in,
                              void* d_out, int out_size, void* d_ws, size_t ws_size,
                              hipStream_t stream) {
  (void)in_sizes; (void)n_in; (void)out_size; (void)ws_size;
  const float* x      = (const float*)d_in[0];
  const float* coords = (const float*)d_in[1];
  float* out = (float*)d_out;
  float* wsP = (float*)d_ws;   // needs 256 * 13120 * 4 B ~= 13.4 MB

  k_colprefix<<<256, 256, 0, stream>>>(x, wsP);
  k_rowprefix<<<256, 256, 0, stream>>>(wsP);
  k_winscores<<<dim3((TOTAL_W + 255) / 256, 256), 256, 0, stream>>>(wsP, out);
  k_nms<<<dim3(256, 3), 256, 0, stream>>>(coords, out);
}